// Vanilla_78520592106142
// MI455X (gfx1250) — compile-verified
//
#include <hip/hip_runtime.h>

typedef __attribute__((ext_vector_type(2))) float v2f;
typedef __attribute__((ext_vector_type(4))) float v4f;
typedef __attribute__((ext_vector_type(8))) float v8f;

#define C_ 3
#define H_ 1024
#define W_ 1024
#define NN_ ((size_t)4096)
#define SZ_A ((size_t)4 * 4096 * 4096)        // floats in A
#define SZ_D ((size_t)4 * 3 * 1024 * 1024)    // floats in data

#define ZBLOCKS 8192    // zero A: 8192 blocks * 256 thr * 8 float4 = 16,777,216 float4
#define CBLOCKS 12288   // copy data: 12288 blocks * 256 thr * 1 float4 = 3,145,728 float4

// ---------------- Kernel 1: zero A (nontemporal b128) + copy data via async-LDS DMA ----------
__global__ __launch_bounds__(256) void zero_copy_kernel(const float* __restrict__ src,
                                                        float* __restrict__ out) {
    const int bid = blockIdx.x;
    const int tid = threadIdx.x;
    if (bid < ZBLOCKS) {
        v4f* o4 = reinterpret_cast<v4f*>(out);
        size_t base = (size_t)bid * 2048 + tid;
        v4f z = {0.f, 0.f, 0.f, 0.f};
#pragma unroll
        for (int k = 0; k < 8; ++k)
            __builtin_nontemporal_store(z, &o4[base + (size_t)k * 256]);
    } else {
        // Async DMA: global -> LDS -> global, tracked by ASYNCcnt (gfx1250 path)
        const int cb = bid - ZBLOCKS;
        __shared__ __align__(16) unsigned char lbuf[4096];
        unsigned ldsAddr = (unsigned)(size_t)(lbuf + tid * 16);          // low 32 bits = LDS offset
        unsigned gOff = ((unsigned)cb * 256u + (unsigned)tid) * 16u;     // byte offset < 2^26
        unsigned long long srcBase = (unsigned long long)(const void*)src;
        unsigned long long dstBase = (unsigned long long)(const void*)(out + SZ_A);
        asm volatile("global_load_async_to_lds_b128 %0, %1, %2"
                     :: "v"(ldsAddr), "v"(gOff), "s"(srcBase) : "memory");
        asm volatile("s_wait_asynccnt 0" ::: "memory");
        asm volatile("global_store_async_from_lds_b128 %0, %1, %2"
                     :: "v"(gOff), "v"(ldsAddr), "s"(dstBase) : "memory");
        asm volatile("s_wait_asynccnt 0" ::: "memory");
    }
}

// ---------------- Kernel 2: edge affinities via V_WMMA_F32_16X16X4_F32 ----------------------
// One wave per task = (dir, b, y, 16-cell chunk). 1008 vertical + 1024 horizontal = 2032 tasks.
// A-matrix: M = cell-in-chunk (16), K = 4 reduction elements; B = ones (row-sum reduction).
// Per (c,i) row: batch all 8 b64 loads into one clause, then 4 WMMAs on 2 independent chains.
__global__ __launch_bounds__(256) void edge_kernel(const float* __restrict__ data,
                                                   float* __restrict__ out) {
    const int wave = threadIdx.x >> 5;
    const int lane = threadIdx.x & 31;
    const int task = blockIdx.x * 8 + wave;

    int dir, b, y, chunk;
    if (task < 1008) {                     // vertical: y in 0..62
        dir = 0; b = task / 252; int r = task % 252; y = r >> 2; chunk = r & 3;
    } else {                               // horizontal: y in 0..63
        int t2 = task - 1008;
        dir = 1; b = t2 >> 8; int r = t2 & 255; y = r >> 2; chunk = r & 3;
    }

    const int m  = lane & 15;              // M (cell) index
    const int hi = lane >> 4;              // 0: K=0,1   1: K=2,3
    const int wBase = chunk * 256 + m * 16;

    const size_t plane = (size_t)H_ * W_;
    const float* pb = data + (size_t)b * C_ * plane;

    v8f acc0 = {0.f, 0.f, 0.f, 0.f, 0.f, 0.f, 0.f, 0.f};
    v8f acc1 = {0.f, 0.f, 0.f, 0.f, 0.f, 0.f, 0.f, 0.f};
    v2f ones = {1.0f, 1.0f};               // B = all-ones: layout-independent

    for (int c = 0; c < C_; ++c) {
        const float* pc = pb + (size_t)c * plane;
        for (int i = 0; i < 15; ++i) {
            const float* row0 = pc + (size_t)(y * 16 + i) * W_;
            const float* row1 = (dir == 0) ? (row0 + 16 * W_)   // cell below (y+1)
                                           : (row0 + 16);       // cell to the right (x+1)
            v2f x0[4], x1[4];
#pragma unroll
            for (int q = 0; q < 4; ++q) {   // one clause of 8 loads, single wait
                x0[q] = *(const v2f*)(row0 + wBase + q * 4 + hi * 2);
                x1[q] = *(const v2f*)(row1 + wBase + q * 4 + hi * 2);
            }
#pragma unroll
            for (int q = 0; q < 4; ++q) {
                v2f a;
                a.x = __builtin_fabsf(x1[q].x - x0[q].x);
                a.y = __builtin_fabsf(x1[q].y - x0[q].y);
                if (q == 3 && hi == 1) a.y = 0.f;   // mask j==15 (patch is 15 wide)
                if (q & 1)
                    acc1 = __builtin_amdgcn_wmma_f32_16x16x4_f32(
                        false, a, false, ones, (short)0, acc1, false, false);
                else
                    acc0 = __builtin_amdgcn_wmma_f32_16x16x4_f32(
                        false, a, false, ones, (short)0, acc0, false, false);
            }
        }
    }

    // Combine the two chains; D layout: VGPR r = row M=r (lanes 0-15) / M=8+r (lanes 16-31).
    v8f acc = acc0 + acc1;
    const size_t outB = (size_t)b * (NN_ * NN_);
#pragma unroll
    for (int r = 0; r < 8; ++r) {
        if (m == 0) {                       // lanes 0 and 16 hold rows 0..7 / 8..15
            const int x = chunk * 16 + hi * 8 + r;
            const float val = acc[r];
            if (dir == 0) {
                size_t n1 = (size_t)(y + 1) * 64 + x;   // lower cell
                size_t n2 = (size_t)y * 64 + x;         // upper cell
                out[outB + n1 * NN_ + n2] = val;
                out[outB + n2 * NN_ + n1] = val;
            } else if (x < 63) {
                size_t n1 = (size_t)y * 64 + x;         // left cell
                size_t n2 = n1 + 1;                     // right cell
                out[outB + n1 * NN_ + n2] = val;
                out[outB + n2 * NN_ + n1] = val;
            }
        }
    }
}

extern "C" void kernel_launch(void* const* d_in, const int* in_sizes, int n_in,
                              void* d_out, int out_size, void* d_ws, size_t ws_size,
                              hipStream_t stream) {
    const float* data = (const float*)d_in[0];
    float* out = (float*)d_out;
    (void)in_sizes; (void)n_in; (void)out_size; (void)d_ws; (void)ws_size;

    // Pass 1: zero 256MB adjacency matrix + DMA-copy the 48MB data passthrough.
    zero_copy_kernel<<<ZBLOCKS + CBLOCKS, 256, 0, stream>>>(data, out);
    // Pass 2: WMMA edge reduction + symmetric scatter (stream order guarantees A is zeroed).
    edge_kernel<<<2032 / 8, 256, 0, stream>>>(data, out);
}